// Attention_33182917329161
// MI455X (gfx1250) — compile-verified
//
#include <hip/hip_runtime.h>

typedef float v2f __attribute__((ext_vector_type(2)));
typedef float v8f __attribute__((ext_vector_type(8)));

#define BATCH   8
#define HEADS   16
#define SEQ     1024
#define DIM     64
#define QTILE   64
#define SSTRIDE 1028   // 1024 + 4 dword pad -> conflict-free M-strided LDS A-fragment reads

__global__ __launch_bounds__(256)
void Attention_33182917329161_kernel(const float* __restrict__ Q,
                                     const float* __restrict__ K,
                                     const float* __restrict__ V,
                                     const int*   __restrict__ M,
                                     float* __restrict__ Out,
                                     float* __restrict__ Attn)
{
    extern __shared__ float sS[];            // [QTILE][SSTRIDE] fp32 scores / probs

    const int bh   = blockIdx.x;             // 0..127 (b*H + h)
    const int b    = bh / HEADS;
    const int qt   = blockIdx.y;             // 0..15  (q tile of 64 rows)
    const int tid  = threadIdx.x;
    const int wave = tid >> 5;               // 0..7 (wave32)
    const int lane = tid & 31;
    const int lo   = lane & 15;
    const int hiH  = lane >> 4;              // 0: K=0,1 half, 1: K=2,3 half

    const int qsub = wave & 3;               // which 16-row q subtile
    const int q0   = qt * QTILE + qsub * 16; // global q base for this wave

    const size_t headBase = (size_t)bh * SEQ * DIM;

    // ---------------- Phase 1: scores = Q K^T / sqrt(D) via v_wmma_f32_16x16x4_f32 ----
    // Preload 16 A-fragments (Q) covering D=64 in K=4 steps.
    v2f aq[16];
    {
        const float* qrow = Q + headBase + (size_t)(q0 + lo) * DIM;
#pragma unroll
        for (int s = 0; s < 16; ++s) {
            const int d = s * 4 + hiH * 2;
            aq[s] = *(const v2f*)(qrow + d);
        }
    }
    const int khalf = wave >> 2;             // waves 0-3: keys 0..511, waves 4-7: 512..1023
    for (int kt = 0; kt < 32; ++kt) {
        const int kbase = khalf * 512 + kt * 16;
        const float* krow = K + headBase + (size_t)(kbase + lo) * DIM;
        v8f acc = {};
#pragma unroll
        for (int s = 0; s < 16; ++s) {
            const int d = s * 4 + hiH * 2;
            v2f bk = *(const v2f*)(krow + d);   // B[d][n] = K[kbase+n][d]
            acc = __builtin_amdgcn_wmma_f32_16x16x4_f32(
                      false, aq[s], false, bk, (short)0, acc, false, false);
        }
#pragma unroll
        for (int r = 0; r < 8; ++r) {
            const int row = qsub * 16 + r + hiH * 8;   // C layout: VGPR r -> M=r / r+8
            sS[row * SSTRIDE + kbase + lo] = acc[r] * 0.125f;  // 1/sqrt(64)
        }
    }
    __syncthreads();

    // ---------------- Phase 2: softmax, then post-softmax mask; write attn ------------
    {
        const int* mbase = M + (size_t)b * SEQ * SEQ;   // mask[B,1,S,S]
        for (int i = 0; i < 8; ++i) {
            const int row = wave * 8 + i;               // 0..63
            const int qg  = qt * QTILE + row;
            float* srow = sS + row * SSTRIDE;

            float mx = -3.402823e38f;
            for (int c = lane; c < SEQ; c += 32) mx = fmaxf(mx, srow[c]);
#pragma unroll
            for (int off = 16; off > 0; off >>= 1) mx = fmaxf(mx, __shfl_xor(mx, off, 32));

            float sum = 0.f;
            for (int c = lane; c < SEQ; c += 32) sum += __expf(srow[c] - mx);
#pragma unroll
            for (int off = 16; off > 0; off >>= 1) sum += __shfl_xor(sum, off, 32);
            const float inv = 1.0f / sum;

            const int*   mrow = mbase + (size_t)qg * SEQ;
            float*       arow = Attn + ((size_t)bh * SEQ + qg) * SEQ;
            for (int c = lane; c < SEQ; c += 32) {
                float p = __expf(srow[c] - mx) * inv;
                p = (mrow[c] == 0) ? -100000.0f : p;    // mask applied AFTER softmax
                srow[c] = p;                            // masked P feeds P@V (faithful)
                arow[c] = p;                            // coalesced attn write
            }
        }
    }
    __syncthreads();

    // ---------------- Phase 3: out = P @ V via v_wmma_f32_16x16x4_f32 -----------------
    {
        const int dbase = (wave >> 2) * 32;             // waves split D: cols 0..31 / 32..63
        const float* vcol = V + headBase + dbase + lo;
        const float* prow = sS + (qsub * 16 + lo) * SSTRIDE;  // A: M = lo
        v8f acc0 = {}, acc1 = {};
        for (int k0 = 0; k0 < SEQ; k0 += 4) {
            const int kk = k0 + hiH * 2;
            v2f ap = *(const v2f*)(prow + kk);          // P[M=lo][kk], P[M=lo][kk+1]
            const float* v0 = vcol + (size_t)kk * DIM;
            v2f b0, b1;
            b0.x = v0[0];        b0.y = v0[DIM];        // V[kk][n], V[kk+1][n]
            b1.x = v0[16];       b1.y = v0[DIM + 16];   // n+16 tile shares A-fragment
            acc0 = __builtin_amdgcn_wmma_f32_16x16x4_f32(
                       false, ap, false, b0, (short)0, acc0, false, false);
            acc1 = __builtin_amdgcn_wmma_f32_16x16x4_f32(
                       false, ap, false, b1, (short)0, acc1, false, false);
        }
#pragma unroll
        for (int r = 0; r < 8; ++r) {
            const int row = qsub * 16 + r + hiH * 8;
            const int qg  = qt * QTILE + row;
            float* orow = Out + ((size_t)bh * SEQ + qg) * DIM + dbase;
            orow[lo]      = acc0[r];
            orow[lo + 16] = acc1[r];
        }
    }
}

extern "C" void kernel_launch(void* const* d_in, const int* in_sizes, int n_in,
                              void* d_out, int out_size, void* d_ws, size_t ws_size,
                              hipStream_t stream) {
    (void)in_sizes; (void)n_in; (void)d_ws; (void)ws_size; (void)out_size;

    const float* Q = (const float*)d_in[0];
    const float* K = (const float*)d_in[1];
    const float* V = (const float*)d_in[2];
    const int*   M = (const int*)d_in[3];

    float* Out  = (float*)d_out;                               // [B,H,S,D]
    float* Attn = Out + (size_t)BATCH * HEADS * SEQ * DIM;     // [B,H,S,S]

    const int ldsBytes = QTILE * SSTRIDE * sizeof(float);      // 263168 B < 320 KB WGP LDS
    hipFuncSetAttribute((const void*)Attention_33182917329161_kernel,
                        hipFuncAttributeMaxDynamicSharedMemorySize, ldsBytes);

    dim3 grid(BATCH * HEADS, SEQ / QTILE, 1);   // 128 x 16 workgroups
    dim3 block(256, 1, 1);                      // 8 wave32s
    Attention_33182917329161_kernel<<<grid, block, ldsBytes, stream>>>(Q, K, V, M, Out, Attn);
}